// FastICABaseline_18511309046064
// MI455X (gfx1250) — compile-verified
//
#include <hip/hip_runtime.h>

typedef __attribute__((ext_vector_type(2))) float v2f;
typedef __attribute__((ext_vector_type(8))) float v8f;

#define C_CH 16
#define T_N 8192
#define NC 4
#define ICA_ITERS 50
#define NTHREADS 256
#define SQRT_T 90.50966799187809f
#define INV_T 1.220703125e-4f

// ---------- serial 4x4 symmetric decorrelation: W <- (W W^T)^{-1/2} W ----------
__device__ void sym_decorr4(const float* Win, float* Wout) {
  float Y[16], V[16];
#pragma unroll
  for (int i = 0; i < 4; ++i)
#pragma unroll
    for (int j = 0; j < 4; ++j) {
      float s = 0.f;
#pragma unroll
      for (int k = 0; k < 4; ++k) s += Win[i * 4 + k] * Win[j * 4 + k];
      Y[i * 4 + j] = s;
    }
#pragma unroll
  for (int i = 0; i < 16; ++i) V[i] = ((i % 5) == 0) ? 1.f : 0.f;
  for (int sweep = 0; sweep < 8; ++sweep) {
    for (int p = 0; p < 3; ++p)
      for (int q = p + 1; q < 4; ++q) {
        float apq = Y[p * 4 + q];
        float app = Y[p * 4 + p], aqq = Y[q * 4 + q];
        if (fabsf(apq) < 1e-12f * (fabsf(app) + fabsf(aqq)) + 1e-30f) continue;
        float tau = (aqq - app) / (2.f * apq);
        float t = ((tau >= 0.f) ? 1.f : -1.f) / (fabsf(tau) + sqrtf(1.f + tau * tau));
        float c = rsqrtf(1.f + t * t);
        float s = t * c;
#pragma unroll
        for (int k = 0; k < 4; ++k) {  // column rotation (A J, V J)
          float yp = Y[k * 4 + p], yq = Y[k * 4 + q];
          Y[k * 4 + p] = c * yp - s * yq;
          Y[k * 4 + q] = s * yp + c * yq;
          float vp = V[k * 4 + p], vq = V[k * 4 + q];
          V[k * 4 + p] = c * vp - s * vq;
          V[k * 4 + q] = s * vp + c * vq;
        }
#pragma unroll
        for (int k = 0; k < 4; ++k) {  // row rotation (J^T A)
          float yp = Y[p * 4 + k], yq = Y[q * 4 + k];
          Y[p * 4 + k] = c * yp - s * yq;
          Y[q * 4 + k] = s * yp + c * yq;
        }
      }
  }
  float isq[4];
#pragma unroll
  for (int k = 0; k < 4; ++k) isq[k] = rsqrtf(fmaxf(Y[k * 4 + k], 1e-24f));
  float M[16];
#pragma unroll
  for (int i = 0; i < 4; ++i)
#pragma unroll
    for (int j = 0; j < 4; ++j) {
      float s = 0.f;
#pragma unroll
      for (int k = 0; k < 4; ++k) s += V[i * 4 + k] * isq[k] * V[j * 4 + k];
      M[i * 4 + j] = s;
    }
#pragma unroll
  for (int i = 0; i < 4; ++i)
#pragma unroll
    for (int j = 0; j < 4; ++j) {
      float s = 0.f;
#pragma unroll
      for (int k = 0; k < 4; ++k) s += M[i * 4 + k] * Win[k * 4 + j];
      Wout[i * 4 + j] = s;
    }
}

// round-robin pairing of 16 elements: 15 rounds x 8 disjoint pairs
__device__ inline void rr_pair(int r, int i, int& p, int& q) {
  int a = (r + i) % 15;
  int b = (i == 0) ? 15 : (r - i + 15) % 15;
  p = (a < b) ? a : b;
  q = (a < b) ? b : a;
}

__global__ __launch_bounds__(NTHREADS) void fastica_kernel(
    const float* __restrict__ xg, const float* __restrict__ w_init,
    float* __restrict__ out) {
  extern __shared__ float smem[];
  float* X1 = smem;                 // 4*8192 = 32768 whitened signal
  float* covw = X1 + 4 * T_N;       // 8*256 = 2048 per-wave Gram partials
  float* red = covw + 2048;         // 20*256 = 5120 reduction scratch
  float* part = red + 20 * NTHREADS;// 32
  float* meanv = part + 32;         // 16
  float* istd = meanv + 16;         // 16
  float* Amat = istd + 16;          // 256 Gram / Jacobi working matrix
  float* Umat = Amat + 256;         // 256 eigenvectors
  float* Kmat = Umat + 256;         // 4*16 whitening matrix
  float* Wm = Kmat + 64;            // 4*4 unmixing matrix

  const int tid = threadIdx.x;
  const int b = blockIdx.x;
  const float* xs = xg + (size_t)b * (C_CH * T_N);

  // ---- Step 1: per-channel mean / 1/(std+eps) ----
  {
    int c = tid >> 4, j = tid & 15;
    float s = 0.f, s2 = 0.f;
    for (int t = j; t < T_N; t += 16) {
      float v = xs[c * T_N + t];
      s += v;
      s2 += v * v;
    }
    red[tid] = s;
    red[NTHREADS + tid] = s2;
  }
  Umat[tid] = ((tid >> 4) == (tid & 15)) ? 1.f : 0.f;  // U = I
  __syncthreads();
  if (tid < 16) {
    float s = 0.f, s2 = 0.f;
    for (int k = 0; k < 16; ++k) {
      s += red[tid * 16 + k];
      s2 += red[NTHREADS + tid * 16 + k];
    }
    float mu = s * INV_T;
    float var = fmaxf(s2 * INV_T - mu * mu, 0.f);
    meanv[tid] = mu;
    istd[tid] = 1.f / (sqrtf(var) + 1e-8f);
  }
  __syncthreads();

  // ---- Step 2: Gram matrix XT*XT^T via chained V_WMMA_F32_16X16X4_F32 ----
  // A-fragment (16x4) and B-fragment (4x16 = A^T) have identical lane layouts:
  // lanes 0-15: row = lane, K = {0,1}; lanes 16-31: row = lane-16, K = {2,3}.
  {
    int wave = tid >> 5, lane = tid & 31;
    int row = lane & 15, kh = lane >> 4;
    float mu = meanv[row], isd = istd[row];
    const float* xrow = xs + row * T_N + wave * 1024 + 2 * kh;
    v8f acc = {};
    for (int t0 = 0; t0 < 1024; t0 += 4) {
      __builtin_prefetch(xrow + t0 + 64, 0, 1);  // global_prefetch_b8
      float2 xv = *(const float2*)(xrow + t0);
      v2f a;
      a[0] = (xv.x - mu) * isd;
      a[1] = (xv.y - mu) * isd;
      acc = __builtin_amdgcn_wmma_f32_16x16x4_f32(false, a, false, a, (short)0,
                                                  acc, false, false);
    }
#pragma unroll
    for (int r = 0; r < 8; ++r)
      covw[wave * 256 + (kh * 8 + r) * 16 + row] = acc[r];
  }
  __syncthreads();
  {
    float s = 0.f;
#pragma unroll
    for (int w = 0; w < 8; ++w) s += covw[w * 256 + tid];
    Amat[tid] = s;
  }
  __syncthreads();

  // ---- Step 3: parallel cyclic Jacobi eigensolve of 16x16 Gram matrix ----
  for (int sweep = 0; sweep < 10; ++sweep) {
    for (int r = 0; r < 15; ++r) {
      if (tid < 8) {
        int p, q;
        rr_pair(r, tid, p, q);
        float apq = Amat[p * 16 + q];
        float app = Amat[p * 17], aqq = Amat[q * 17];
        float c = 1.f, s = 0.f;
        if (fabsf(apq) > 1e-12f * (fabsf(app) + fabsf(aqq)) + 1e-30f) {
          float tau = (aqq - app) / (2.f * apq);
          float t = ((tau >= 0.f) ? 1.f : -1.f) /
                    (fabsf(tau) + sqrtf(1.f + tau * tau));
          c = rsqrtf(1.f + t * t);
          s = t * c;
        }
        part[tid] = c;
        part[8 + tid] = s;
      }
      __syncthreads();
      {
        int pi = (tid >> 4) & 7;
        int rw = tid & 15;
        int p, q;
        rr_pair(r, pi, p, q);
        float c = part[pi], s = part[8 + pi];
        if (tid < 128) {  // column rotation of A
          float ap = Amat[rw * 16 + p], aq = Amat[rw * 16 + q];
          Amat[rw * 16 + p] = c * ap - s * aq;
          Amat[rw * 16 + q] = s * ap + c * aq;
        } else {  // column rotation of U (V <- V J)
          float up = Umat[rw * 16 + p], uq = Umat[rw * 16 + q];
          Umat[rw * 16 + p] = c * up - s * uq;
          Umat[rw * 16 + q] = s * up + c * uq;
        }
      }
      __syncthreads();
      if (tid < 128) {  // row rotation of A
        int pi = tid >> 4;
        int cl = tid & 15;
        int p, q;
        rr_pair(r, pi, p, q);
        float c = part[pi], s = part[8 + pi];
        float ap = Amat[p * 16 + cl], aq = Amat[q * 16 + cl];
        Amat[p * 16 + cl] = c * ap - s * aq;
        Amat[q * 16 + cl] = s * ap + c * aq;
      }
      __syncthreads();
    }
  }

  // ---- top-4 eigenpairs (descending) -> whitening K; W0 = decorr(w_init) ----
  if (tid == 0) {
    float ev[16];
#pragma unroll
    for (int i = 0; i < 16; ++i) ev[i] = Amat[i * 17];
    unsigned usedmask = 0;
    for (int j = 0; j < NC; ++j) {
      int best = 0;
      float bv = -1e30f;
      for (int i = 0; i < 16; ++i)
        if (!((usedmask >> i) & 1) && ev[i] > bv) {
          bv = ev[i];
          best = i;
        }
      usedmask |= 1u << best;
      float dinv = rsqrtf(fmaxf(bv, 1e-24f));  // 1/d, d = sqrt(eigenvalue)
      float sg = (Umat[best] >= 0.f) ? 1.f : -1.f;  // sklearn sign(u[0])
      for (int c = 0; c < 16; ++c) Kmat[j * 16 + c] = sg * Umat[c * 16 + best] * dinv;
    }
    float W0in[16], W0[16];
#pragma unroll
    for (int i = 0; i < 16; ++i) W0in[i] = w_init[i];
    sym_decorr4(W0in, W0);
#pragma unroll
    for (int i = 0; i < 16; ++i) Wm[i] = W0[i];
  }
  __syncthreads();

  // ---- Step 4: X1 = (K @ XT) * sqrt(T) resident in LDS ----
  for (int t = tid; t < T_N; t += NTHREADS) {
    float s0 = 0.f, s1 = 0.f, s2 = 0.f, s3 = 0.f;
#pragma unroll
    for (int c = 0; c < 16; ++c) {
      float xv = (xs[c * T_N + t] - meanv[c]) * istd[c];
      s0 += Kmat[c] * xv;
      s1 += Kmat[16 + c] * xv;
      s2 += Kmat[32 + c] * xv;
      s3 += Kmat[48 + c] * xv;
    }
    X1[t] = s0 * SQRT_T;
    X1[T_N + t] = s1 * SQRT_T;
    X1[2 * T_N + t] = s2 * SQRT_T;
    X1[3 * T_N + t] = s3 * SQRT_T;
  }
  __syncthreads();

  // ---- Step 5: 50 fixed-point iterations, logcosh nonlinearity ----
  for (int it = 0; it < ICA_ITERS; ++it) {
    float W[16];
#pragma unroll
    for (int i = 0; i < 16; ++i) W[i] = Wm[i];
    float acc[16], gsq[4];
#pragma unroll
    for (int i = 0; i < 16; ++i) acc[i] = 0.f;
#pragma unroll
    for (int i = 0; i < 4; ++i) gsq[i] = 0.f;
    for (int t = tid; t < T_N; t += NTHREADS) {
      float x0 = X1[t], x1 = X1[T_N + t], x2 = X1[2 * T_N + t], x3 = X1[3 * T_N + t];
      float y0 = W[0] * x0 + W[1] * x1 + W[2] * x2 + W[3] * x3;
      float y1 = W[4] * x0 + W[5] * x1 + W[6] * x2 + W[7] * x3;
      float y2 = W[8] * x0 + W[9] * x1 + W[10] * x2 + W[11] * x3;
      float y3 = W[12] * x0 + W[13] * x1 + W[14] * x2 + W[15] * x3;
      float g0, g1, g2, g3;
      asm volatile(
          "v_tanh_f32 %0, %4\n\t"
          "v_tanh_f32 %1, %5\n\t"
          "v_tanh_f32 %2, %6\n\t"
          "v_tanh_f32 %3, %7\n\t"
          "s_nop 1"
          : "=&v"(g0), "=&v"(g1), "=&v"(g2), "=&v"(g3)
          : "v"(y0), "v"(y1), "v"(y2), "v"(y3));
      gsq[0] += g0 * g0;
      gsq[1] += g1 * g1;
      gsq[2] += g2 * g2;
      gsq[3] += g3 * g3;
      acc[0] += g0 * x0; acc[1] += g0 * x1; acc[2] += g0 * x2; acc[3] += g0 * x3;
      acc[4] += g1 * x0; acc[5] += g1 * x1; acc[6] += g1 * x2; acc[7] += g1 * x3;
      acc[8] += g2 * x0; acc[9] += g2 * x1; acc[10] += g2 * x2; acc[11] += g2 * x3;
      acc[12] += g3 * x0; acc[13] += g3 * x1; acc[14] += g3 * x2; acc[15] += g3 * x3;
    }
#pragma unroll
    for (int k = 0; k < 16; ++k) red[k * NTHREADS + tid] = acc[k];
#pragma unroll
    for (int k = 0; k < 4; ++k) red[(16 + k) * NTHREADS + tid] = gsq[k];
    __syncthreads();
    if (tid < 20) {
      float s = 0.f;
      for (int i = 0; i < NTHREADS; ++i) s += red[tid * NTHREADS + i];
      part[tid] = s;
    }
    __syncthreads();
    if (tid == 0) {
      float W1[16], Wn[16];
#pragma unroll
      for (int i = 0; i < 4; ++i) {
        float gw = 1.f - part[16 + i] * INV_T;  // mean(1 - g^2)
#pragma unroll
        for (int j = 0; j < 4; ++j)
          W1[i * 4 + j] = part[i * 4 + j] * INV_T - gw * Wm[i * 4 + j];
      }
      sym_decorr4(W1, Wn);
#pragma unroll
      for (int i = 0; i < 16; ++i) Wm[i] = Wn[i];
    }
    __syncthreads();
  }

  // ---- Step 6: S = W @ X1, unit-variance scaling, store ----
  {
    float W[16];
#pragma unroll
    for (int i = 0; i < 16; ++i) W[i] = Wm[i];
    float sm[4] = {0.f, 0.f, 0.f, 0.f}, ss[4] = {0.f, 0.f, 0.f, 0.f};
    for (int t = tid; t < T_N; t += NTHREADS) {
      float x0 = X1[t], x1 = X1[T_N + t], x2 = X1[2 * T_N + t], x3 = X1[3 * T_N + t];
      float y0 = W[0] * x0 + W[1] * x1 + W[2] * x2 + W[3] * x3;
      float y1 = W[4] * x0 + W[5] * x1 + W[6] * x2 + W[7] * x3;
      float y2 = W[8] * x0 + W[9] * x1 + W[10] * x2 + W[11] * x3;
      float y3 = W[12] * x0 + W[13] * x1 + W[14] * x2 + W[15] * x3;
      sm[0] += y0; ss[0] += y0 * y0;
      sm[1] += y1; ss[1] += y1 * y1;
      sm[2] += y2; ss[2] += y2 * y2;
      sm[3] += y3; ss[3] += y3 * y3;
    }
#pragma unroll
    for (int k = 0; k < 4; ++k) {
      red[k * NTHREADS + tid] = sm[k];
      red[(4 + k) * NTHREADS + tid] = ss[k];
    }
    __syncthreads();
    if (tid < 8) {
      float s = 0.f;
      for (int i = 0; i < NTHREADS; ++i) s += red[tid * NTHREADS + i];
      part[tid] = s;
    }
    __syncthreads();
    float inv[4];
#pragma unroll
    for (int k = 0; k < 4; ++k) {
      float mu = part[k] * INV_T;
      float var = fmaxf(part[4 + k] * INV_T - mu * mu, 1e-30f);
      inv[k] = rsqrtf(var);
    }
    float* od = out + (size_t)b * (NC * T_N);
    for (int t = tid; t < T_N; t += NTHREADS) {
      float x0 = X1[t], x1 = X1[T_N + t], x2 = X1[2 * T_N + t], x3 = X1[3 * T_N + t];
      float y0 = W[0] * x0 + W[1] * x1 + W[2] * x2 + W[3] * x3;
      float y1 = W[4] * x0 + W[5] * x1 + W[6] * x2 + W[7] * x3;
      float y2 = W[8] * x0 + W[9] * x1 + W[10] * x2 + W[11] * x3;
      float y3 = W[12] * x0 + W[13] * x1 + W[14] * x2 + W[15] * x3;
      od[t] = y0 * inv[0];
      od[T_N + t] = y1 * inv[1];
      od[2 * T_N + t] = y2 * inv[2];
      od[3 * T_N + t] = y3 * inv[3];
    }
  }
}

extern "C" void kernel_launch(void* const* d_in, const int* in_sizes, int n_in,
                              void* d_out, int out_size, void* d_ws,
                              size_t ws_size, hipStream_t stream) {
  const float* x = (const float*)d_in[0];
  const float* w_init = (const float*)d_in[1];
  float* out = (float*)d_out;
  int B = in_sizes[0] / (C_CH * T_N);
  constexpr size_t SMEM =
      (4 * T_N + 2048 + 20 * NTHREADS + 32 + 16 + 16 + 256 + 256 + 64 + 16) *
      sizeof(float);
  hipFuncSetAttribute(reinterpret_cast<const void*>(fastica_kernel),
                      hipFuncAttributeMaxDynamicSharedMemorySize, (int)SMEM);
  fastica_kernel<<<dim3(B), dim3(NTHREADS), SMEM, stream>>>(x, w_init, out);
  (void)n_in;
  (void)out_size;
  (void)d_ws;
  (void)ws_size;
}